// FuzzyMergeLayer_20547123544800
// MI455X (gfx1250) — compile-verified
//
#include <hip/hip_runtime.h>

#define HW    65536          // H*W
#define CCH   16             // channels
#define KCL   8              // classes
#define BATCH 8
#define KBHW  (BATCH * HW)   // 524288: per-class stride of gauss scratch

typedef __attribute__((ext_vector_type(2))) float v2f;
typedef __attribute__((ext_vector_type(4))) float v4f;
typedef __attribute__((ext_vector_type(8))) float v8f;

// ---------------------------------------------------------------------------
// Kernel 0: per-class 16x16 lower-triangular inverse (forward substitution,
// one thread per (class, column)) + logdet.  Tiny, runs once per launch.
// ---------------------------------------------------------------------------
__global__ __launch_bounds__(128) void prep_kernel(const float* __restrict__ scale,
                                                   float* __restrict__ Linv,
                                                   float* __restrict__ logdet) {
    const int t   = threadIdx.x;        // 0..127
    const int k   = t >> 4;
    const int col = t & 15;
    const float* Lk = scale + k * CCH * CCH;   // scale[k,0,:,:], use lower tri only
    float y[CCH];
#pragma unroll
    for (int i = 0; i < CCH; ++i) {
        float s = (i == col) ? 1.0f : 0.0f;
#pragma unroll
        for (int j = 0; j < CCH; ++j) {
            if (j < i) s -= Lk[i * CCH + j] * y[j];
        }
        y[i] = s / Lk[i * CCH + i];            // rows above col solve to exact 0
        Linv[k * CCH * CCH + i * CCH + col] = y[i];
    }
    if (col == 0) {
        float ld = 0.0f;
#pragma unroll
        for (int i = 0; i < CCH; ++i) ld += __logf(fabsf(Lk[i * CCH + i]));
        logdet[k] = ld;
    }
}

// ---------------------------------------------------------------------------
// Kernel 1: WMMA gaussian density.  Each wave owns a 16-pixel tile:
//   z(16x16) = Linv[k](16x16) x diff(16ch x 16px) via 4x V_WMMA_F32_16X16X4_F32
// Writes gauss in class-major [K, B*HW] layout so the reshape scramble in the
// finalize pass becomes a contiguous read.
// ---------------------------------------------------------------------------
__global__ __launch_bounds__(256) void gauss_kernel(const float* __restrict__ x,
                                                    const float* __restrict__ mean,
                                                    const float* __restrict__ Linv,
                                                    const float* __restrict__ logdet,
                                                    float* __restrict__ gauss) {
    const int  lane = threadIdx.x & 31;
    const int  wave = threadIdx.x >> 5;
    const int  n    = lane & 15;          // pixel column within tile / A row
    const int  hi   = lane >> 4;          // half-wave select
    const long tile = (long)blockIdx.x * 8 + wave;
    const long P    = tile * 16;          // base pixel, flat over BATCH*HW

    // Load the 16px x 16ch tile of x directly in WMMA-B layout:
    // lane(n,hi) step t holds channels {4t+2hi, 4t+2hi+1} of pixel n.
    v2f xv[4];
    const float* xp = x + (P + n) * CCH + hi * 2;
#pragma unroll
    for (int t = 0; t < 4; ++t)
        xv[t] = *(const v2f*)(xp + t * 4);

#pragma unroll
    for (int k = 0; k < KCL; ++k) {
        const float* Ak = Linv + k * CCH * CCH + n * CCH + hi * 2;
        const float* mk = mean + k * CCH + hi * 2;
        v8f acc = {0.f, 0.f, 0.f, 0.f, 0.f, 0.f, 0.f, 0.f};
#pragma unroll
        for (int t = 0; t < 4; ++t) {
            v2f a = *(const v2f*)(Ak + t * 4);   // A: Linv[n][4t+2hi .. +1]
            v2f m = *(const v2f*)(mk + t * 4);
            v2f b = xv[t] - m;                   // B: diff[4t+2hi..+1][n]
            acc = __builtin_amdgcn_wmma_f32_16x16x4_f32(
                false, a, false, b, (short)0, acc, false, false);
        }
        // lane holds z[8hi .. 8hi+7][n]; combine halves for Mahalanobis dist
        float partial = 0.f;
#pragma unroll
        for (int r = 0; r < 8; ++r) partial += acc[r] * acc[r];
        float maha = partial + __shfl_xor(partial, 16, 32);
        // logp = -0.5*maha - 0.5*C*log(2pi) - logdet;  0.5*16*log(2pi)=14.70301653
        float g = __expf(-0.5f * maha - 14.7030165312747f - logdet[k]);
        if (hi == 0)
            gauss[(long)k * KBHW + P + n] = g;
    }
}

// ---------------------------------------------------------------------------
// Kernel 2: bandwidth pass.  For output group (n, p):
//   probs kk=0..7 come from gauss flat [n*KBHW + p*8 + kk] (contiguous 32B),
//   L2-normalize that group, concat with x[n, p, 0:16] -> out row of 24 f32.
// ---------------------------------------------------------------------------
__global__ __launch_bounds__(256) void finalize_kernel(const float* __restrict__ x,
                                                       const float* __restrict__ gauss,
                                                       float* __restrict__ out) {
    const long idx = (long)blockIdx.x * 256 + threadIdx.x;   // 0 .. 8*HW-1
    const int  n   = (int)(idx >> 16);                       // leading dim (=K)
    const long gb  = (long)n * KBHW + (idx & 0xFFFF) * 8;

    v4f g0 = *(const v4f*)(gauss + gb);
    v4f g1 = *(const v4f*)(gauss + gb + 4);
    float ss = g0.x * g0.x + g0.y * g0.y + g0.z * g0.z + g0.w * g0.w +
               g1.x * g1.x + g1.y * g1.y + g1.z * g1.z + g1.w * g1.w;
    float inv = rsqrtf(fmaxf(ss, 1e-12f));

    const v4f* xp = (const v4f*)(x + idx * 16);
    v4f*       op = (v4f*)(out + idx * 24);
    op[0] = xp[0];
    op[1] = xp[1];
    op[2] = xp[2];
    op[3] = xp[3];
    op[4] = g0 * inv;
    op[5] = g1 * inv;
}

// ---------------------------------------------------------------------------
extern "C" void kernel_launch(void* const* d_in, const int* in_sizes, int n_in,
                              void* d_out, int out_size, void* d_ws, size_t ws_size,
                              hipStream_t stream) {
    const float* x     = (const float*)d_in[0];   // [B,H,W,C]
    const float* scale = (const float*)d_in[1];   // [K,1,C,C]
    const float* mean  = (const float*)d_in[2];   // [K,1,C]
    float*       out   = (float*)d_out;           // [8,H,W,C+K]

    float* ws     = (float*)d_ws;
    float* gauss  = ws;                                // K*B*HW floats (16.8 MB)
    float* Linv   = ws + (long)KCL * KBHW;             // K*C*C floats
    float* logdet = Linv + KCL * CCH * CCH;            // K floats

    prep_kernel<<<1, 128, 0, stream>>>(scale, Linv, logdet);
    // BATCH*HW / (16 px/wave * 8 waves/block) = 4096 blocks
    gauss_kernel<<<KBHW / 128, 256, 0, stream>>>(x, mean, Linv, logdet, gauss);
    // 8*HW output groups / 256 threads = 2048 blocks
    finalize_kernel<<<(KCL * HW) / 256, 256, 0, stream>>>(x, gauss, out);
}